// BackboneNet_18923625906314
// MI455X (gfx1250) — compile-verified
//
#include <hip/hip_runtime.h>
#include <math.h>

typedef __attribute__((ext_vector_type(2))) float v2f;
typedef __attribute__((ext_vector_type(8))) float v8f;

#define IN_DIM 256
#define OUT_DIM 256   // N_HEADS * HEAD_DIM
#define N_HEADS 4
#define HEAD_DIM 64
#define NEG_SLOPE 0.2f

// ---------------------------------------------------------------------------
// Kernel 1: xp = x @ W   ([N,256] x [256,256] -> [N,256]) via fp32 WMMA.
// One wave per 16x16 output tile; 64 x V_WMMA_F32_16X16X4_F32 over K=256.
// ---------------------------------------------------------------------------
__global__ __launch_bounds__(256) void gat_gemm_xp(const float* __restrict__ X,
                                                   const float* __restrict__ W,
                                                   float* __restrict__ XP, int N) {
    const int wave = (blockIdx.x * blockDim.x + threadIdx.x) >> 5;
    const int lane = threadIdx.x & 31;

    const int tiles_n = OUT_DIM / 16;           // 16 column tiles
    const int tile_m = wave / tiles_n;
    const int tile_n = wave - tile_m * tiles_n;
    const int m0 = tile_m * 16;
    if (m0 >= N) return;
    const int n0 = tile_n * 16;

    const int l15   = lane & 15;
    const int khalf = (lane >> 4) * 2;          // 0 or 2
    const int row   = min(m0 + l15, N - 1);     // A: M = lane&15
    const int col   = n0 + l15;                 // B: N = lane&15

    v8f acc = {};
    const float* __restrict__ xrow = X + (size_t)row * IN_DIM + khalf;
    const float* __restrict__ wcol = W + (size_t)khalf * OUT_DIM + col;

#pragma unroll 8
    for (int k0 = 0; k0 < IN_DIM; k0 += 4) {
        // A fragment: (M=lane&15, K=k0+khalf+{0,1})  -- 8B aligned -> global_load_b64
        v2f a = *(const v2f*)(xrow + k0);
        // B fragment: (K=k0+khalf+{0,1}, N=lane&15)
        v2f b;
        b.x = wcol[(size_t)k0 * OUT_DIM];
        b.y = wcol[(size_t)k0 * OUT_DIM + OUT_DIM];
        acc = __builtin_amdgcn_wmma_f32_16x16x4_f32(
            /*neg_a=*/false, a, /*neg_b=*/false, b,
            /*c_mod=*/(short)0, acc, /*reuse_a=*/false, /*reuse_b=*/false);
    }

    // D layout: VGPR r -> row m0 + r + 8*(lane>=16), col n0 + (lane&15)
    const int rbase = m0 + (lane >> 4) * 8;
    float* __restrict__ outp = XP + (size_t)rbase * OUT_DIM + col;
    if (m0 + 16 <= N) {
        // Full tile: branch-free store clause, immediate offsets r*1024B.
#pragma unroll
        for (int r = 0; r < 8; ++r) outp[r * OUT_DIM] = acc[r];
    } else {
        // Final partial tile only.
#pragma unroll
        for (int r = 0; r < 8; ++r)
            if (rbase + r < N) outp[r * OUT_DIM] = acc[r];
    }
}

// ---------------------------------------------------------------------------
// Kernel 2: a_src[n,h] = dot(xp[n,h,:], att_src[h,:]); same for a_dst.
// One wave per node; lane owns 8 consecutive channels (8 lanes per head).
// ---------------------------------------------------------------------------
__global__ __launch_bounds__(256) void gat_attn_dots(const float* __restrict__ XP,
                                                     const float* __restrict__ att_src,
                                                     const float* __restrict__ att_dst,
                                                     float* __restrict__ a_src,
                                                     float* __restrict__ a_dst, int N) {
    const int node = (blockIdx.x * blockDim.x + threadIdx.x) >> 5;
    const int lane = threadIdx.x & 31;
    if (node >= N) return;

    const float* xp = XP + (size_t)node * OUT_DIM;
    const int base = lane * 8;                 // channels [base, base+8)
    const int head = base >> 6;                // head = lane/8

    float ps = 0.f, pd = 0.f;
#pragma unroll
    for (int j = 0; j < 8; ++j) {
        float v = xp[base + j];
        ps = fmaf(v, att_src[base + j], ps);   // att_* flattened [H*C] == [256]
        pd = fmaf(v, att_dst[base + j], pd);
    }
#pragma unroll
    for (int off = 1; off < 8; off <<= 1) {
        ps += __shfl_xor(ps, off, 32);
        pd += __shfl_xor(pd, off, 32);
    }
    if ((lane & 7) == 0) {
        a_src[node * N_HEADS + head] = ps;
        a_dst[node * N_HEADS + head] = pd;
    }
}

// ---------------------------------------------------------------------------
// Kernel 3: init out=0, e_max=-inf, denom=0.
// ---------------------------------------------------------------------------
__global__ void gat_init(float* __restrict__ out, float* __restrict__ e_max,
                         float* __restrict__ denom, int N) {
    int i = blockIdx.x * blockDim.x + threadIdx.x;
    if (i < N * OUT_DIM) out[i] = 0.f;
    if (i < N * N_HEADS) {
        e_max[i] = -__builtin_inff();
        denom[i] = 0.f;
    }
}

// fp32 atomic max via sign-split int-max / uint-min (no CAS loop).
__device__ inline void atomicMaxF32(float* addr, float v) {
    if (v >= 0.f) atomicMax((int*)addr, __float_as_int(v));
    else          atomicMin((unsigned int*)addr, (unsigned int)__float_as_int(v));
}

// ---------------------------------------------------------------------------
// Kernel 4: per-edge logits e = leaky_relu(a_src[src]+a_dst[dst]); seg-max.
// Edges [0,E) are real; [E, E+N) are self-loops.
// ---------------------------------------------------------------------------
__global__ void gat_edge_logits(const long long* __restrict__ ei,
                                const float* __restrict__ a_src,
                                const float* __restrict__ a_dst,
                                float* __restrict__ e_buf,
                                float* __restrict__ e_max, int E, int N) {
    int e = blockIdx.x * blockDim.x + threadIdx.x;
    if (e >= E + N) return;
    int s, d;
    if (e < E) { s = (int)ei[e]; d = (int)ei[(size_t)E + e]; }
    else       { s = d = e - E; }
#pragma unroll
    for (int h = 0; h < N_HEADS; ++h) {
        float v = a_src[s * N_HEADS + h] + a_dst[d * N_HEADS + h];
        v = (v >= 0.f) ? v : NEG_SLOPE * v;
        e_buf[(size_t)e * N_HEADS + h] = v;
        atomicMaxF32(&e_max[d * N_HEADS + h], v);
    }
}

// ---------------------------------------------------------------------------
// Kernel 5: e_exp = exp(e - e_max[dst]); seg-sum into denom. Overwrites e_buf.
// ---------------------------------------------------------------------------
__global__ void gat_edge_exp(const long long* __restrict__ ei,
                             float* __restrict__ e_buf,
                             const float* __restrict__ e_max,
                             float* __restrict__ denom, int E, int N) {
    int e = blockIdx.x * blockDim.x + threadIdx.x;
    if (e >= E + N) return;
    int d = (e < E) ? (int)ei[(size_t)E + e] : (e - E);
#pragma unroll
    for (int h = 0; h < N_HEADS; ++h) {
        float x = __expf(e_buf[(size_t)e * N_HEADS + h] - e_max[d * N_HEADS + h]);
        e_buf[(size_t)e * N_HEADS + h] = x;
        atomicAdd(&denom[d * N_HEADS + h], x);
    }
}

// ---------------------------------------------------------------------------
// Kernel 6: out[dst] += (e_exp/denom[dst]) * xp[src]  (per head channel group)
// One 256-thread block per edge; thread c owns channel c (head = c>>6).
// Coalesced 1KB gather + 256 native global_atomic_add_f32 per edge.
// ---------------------------------------------------------------------------
__global__ __launch_bounds__(256) void gat_edge_aggregate(const long long* __restrict__ ei,
                                                          const float* __restrict__ XP,
                                                          const float* __restrict__ e_buf,
                                                          const float* __restrict__ denom,
                                                          float* __restrict__ out, int E, int N) {
    int e = blockIdx.x;
    int c = threadIdx.x;
    int s, d;
    if (e < E) { s = (int)ei[e]; d = (int)ei[(size_t)E + e]; }
    else       { s = d = e - E; }
    int h = c >> 6;
    float alpha = e_buf[(size_t)e * N_HEADS + h] / denom[d * N_HEADS + h];
    atomicAdd(&out[(size_t)d * OUT_DIM + c], alpha * XP[(size_t)s * OUT_DIM + c]);
}

// ---------------------------------------------------------------------------
// Kernel 7: final ReLU.
// ---------------------------------------------------------------------------
__global__ void gat_relu(float* __restrict__ out, int n) {
    int i = blockIdx.x * blockDim.x + threadIdx.x;
    if (i < n) out[i] = fmaxf(out[i], 0.f);
}

// ---------------------------------------------------------------------------
extern "C" void kernel_launch(void* const* d_in, const int* in_sizes, int n_in,
                              void* d_out, int out_size, void* d_ws, size_t ws_size,
                              hipStream_t stream) {
    const float*     x       = (const float*)d_in[0];
    const long long* ei      = (const long long*)d_in[1];   // int64 [2, E]
    const float*     W       = (const float*)d_in[2];
    const float*     att_src = (const float*)d_in[3];
    const float*     att_dst = (const float*)d_in[4];
    float*           out     = (float*)d_out;

    const int N = in_sizes[0] / IN_DIM;   // 50000
    const int E = in_sizes[1] / 2;        // 800000
    const int ET = E + N;                 // with self-loops

    // Workspace layout (~68 MB)
    char* ws = (char*)d_ws;
    float* XP    = (float*)ws;  ws += (size_t)N * OUT_DIM  * sizeof(float);
    float* a_src = (float*)ws;  ws += (size_t)N * N_HEADS  * sizeof(float);
    float* a_dst = (float*)ws;  ws += (size_t)N * N_HEADS  * sizeof(float);
    float* e_max = (float*)ws;  ws += (size_t)N * N_HEADS  * sizeof(float);
    float* denom = (float*)ws;  ws += (size_t)N * N_HEADS  * sizeof(float);
    float* e_buf = (float*)ws;  ws += (size_t)ET * N_HEADS * sizeof(float);

    // 1) projection GEMM (fp32 WMMA)
    {
        int tiles_m = (N + 15) / 16;
        int waves   = tiles_m * (OUT_DIM / 16);
        int blocks  = (waves * 32 + 255) / 256;
        gat_gemm_xp<<<blocks, 256, 0, stream>>>(x, W, XP, N);
    }
    // 2) attention dot products
    {
        int blocks = (N * 32 + 255) / 256;
        gat_attn_dots<<<blocks, 256, 0, stream>>>(XP, att_src, att_dst, a_src, a_dst, N);
    }
    // 3) init accumulators
    {
        int total = N * OUT_DIM;
        gat_init<<<(total + 255) / 256, 256, 0, stream>>>(out, e_max, denom, N);
    }
    // 4) logits + segment max
    gat_edge_logits<<<(ET + 255) / 256, 256, 0, stream>>>(ei, a_src, a_dst, e_buf, e_max, E, N);
    // 5) exp + segment sum
    gat_edge_exp<<<(ET + 255) / 256, 256, 0, stream>>>(ei, e_buf, e_max, denom, E, N);
    // 6) weighted scatter-aggregate (dominant: ~1.7 GB traffic)
    gat_edge_aggregate<<<ET, 256, 0, stream>>>(ei, XP, e_buf, denom, out, E, N);
    // 7) ReLU
    gat_relu<<<(N * OUT_DIM + 255) / 256, 256, 0, stream>>>(out, N * OUT_DIM);
}